// MACEEdgeMessageBlock_40724879901207
// MI455X (gfx1250) — compile-verified
//
#include <hip/hip_runtime.h>

// ---------------------------------------------------------------------------
// MACE edge-message block for MI455X (gfx1250), fp32 via V_WMMA_F32_16X16X4_F32
// Kernel 1: node up-projection  (node_feats -> ws_node, 20000 x 512, L2-resident)
// Kernel 2: fused per-edge pipeline (MLP -> gather -> tensor product -> out GEMM)
// ---------------------------------------------------------------------------

typedef float v2f __attribute__((ext_vector_type(2)));
typedef float v8f __attribute__((ext_vector_type(8)));

static __device__ __forceinline__ v8f wmma_f32(v2f a, v2f b, v8f c) {
  // D = A(16x4) * B(4x16) + C(16x16), full fp32
  return __builtin_amdgcn_wmma_f32_16x16x4_f32(false, a, false, b, (short)0, c,
                                               false, false);
}

static __device__ __forceinline__ float act_silu(float x) {
  // silu(x) * SILU_NORM
  return 1.6790390826f * x / (1.0f + expf(-x));
}

#define NNODES 20000
#define NEDGES 100000

// ---------------------------------------------------------------------------
// Kernel 1: s = nf[:, :128] @ Wus / sqrt(128); v_i = nf[:,128+3u+i] @ Wuv / sqrt(128)
// ws layout per node (512 f32): [ s(128) | v0(128) | v1(128) | v2(128) ]
// One wave per 16-node tile.
// ---------------------------------------------------------------------------
__global__ __launch_bounds__(256) void mace_node_up(
    const float* __restrict__ nf, const float* __restrict__ Wus,
    const float* __restrict__ Wuv, float* __restrict__ wsn) {
  const int wave = (blockIdx.x * blockDim.x + threadIdx.x) >> 5;
  if (wave >= (NNODES / 16)) return;  // uniform per wave
  const int lane  = threadIdx.x & 31;
  const int halfw = lane >> 4;
  const int l16   = lane & 15;
  const int koff  = halfw * 2;
  const int nbase = wave * 16;
  const float* nfr = nf + (nbase + l16) * 512;
  const v8f vzero = {0.f, 0.f, 0.f, 0.f, 0.f, 0.f, 0.f, 0.f};

  for (int comp = 0; comp < 4; ++comp) {
    // A fragments for the whole K=128 reduction (reused across 8 N-tiles)
    v2f afr[32];
    if (comp == 0) {
#pragma unroll
      for (int k = 0; k < 32; ++k) {
        const int kk = k * 4 + koff;
        v2f a = {nfr[kk], nfr[kk + 1]};
        afr[k] = a;
      }
    } else {
      const int i = comp - 1;
#pragma unroll
      for (int k = 0; k < 32; ++k) {
        const int kk = k * 4 + koff;
        const int c0 = 128 + kk * 3 + i;  // v[n,u,i] = nf[n, 128 + 3u + i]
        v2f a = {nfr[c0], nfr[c0 + 3]};
        afr[k] = a;
      }
    }
    const float* W = (comp == 0) ? Wus : Wuv;
    for (int ng = 0; ng < 2; ++ng) {  // 2 groups of 4 N-tiles => N=128
      v8f acc[4] = {vzero, vzero, vzero, vzero};
#pragma unroll 4
      for (int k = 0; k < 32; ++k) {
        const int kk = k * 4 + koff;
#pragma unroll
        for (int t = 0; t < 4; ++t) {  // 4 independent WMMA chains
          const int n = (ng * 4 + t) * 16 + l16;
          v2f b = {W[kk * 128 + n], W[(kk + 1) * 128 + n]};
          acc[t] = wmma_f32(afr[k], b, acc[t]);
        }
      }
#pragma unroll
      for (int t = 0; t < 4; ++t) {
        const int n = (ng * 4 + t) * 16 + l16;
#pragma unroll
        for (int j = 0; j < 8; ++j) {
          const int r = nbase + j + halfw * 8;  // D: lane half picks M+8
          wsn[r * 512 + comp * 128 + n] = acc[t][j] * 0.08838834764831845f;
        }
      }
    }
  }
}

// ---------------------------------------------------------------------------
// Kernel 2: fused edge pipeline. One wave (32 lanes) per 16-edge tile.
// LDS: hA/hB 16x68 ping-pong, mid 16x1028 (s|v0|v1|v2, 256 each), edge scalars.
// ---------------------------------------------------------------------------
__global__ __launch_bounds__(32) void mace_edge(
    const float* __restrict__ ea, const float* __restrict__ ef,
    const int* __restrict__ snd, const float* __restrict__ w0,
    const float* __restrict__ w1, const float* __restrict__ w2,
    const float* __restrict__ w3, const float* __restrict__ Wos,
    const float* __restrict__ Wov, const float* __restrict__ wsn,
    float* __restrict__ out) {
  extern __shared__ float smem[];
  constexpr int HS = 68;    // 68 % 64 == 4 -> conflict-free b64 row access
  constexpr int MS = 1028;  // 1028 % 64 == 4
  float* hA  = smem;
  float* hB  = smem + 16 * HS;
  float* mid = smem + 32 * HS;
  float* ssh = mid + 16 * MS;       // 16 x 4 edge_attrs
  int*   ssd = (int*)(ssh + 64);    // 16 sender ids

  const int tile  = blockIdx.x;
  const int lane  = threadIdx.x;
  const int halfw = lane >> 4;
  const int l16   = lane & 15;
  const int koff  = halfw * 2;
  const int ebase = tile * 16;
  const v8f vzero = {0.f, 0.f, 0.f, 0.f, 0.f, 0.f, 0.f, 0.f};

  if (lane < 16) {
    const int e = ebase + lane;
    ssd[lane] = snd[e];  // sender = edge_index[0]
    const float4 a = *(const float4*)(ea + e * 4);
    ssh[lane * 4 + 0] = a.x;
    ssh[lane * 4 + 1] = a.y;
    ssh[lane * 4 + 2] = a.z;
    ssh[lane * 4 + 3] = a.w;
  }

  // ---- MLP layer 0: (16x8) @ (8x64), scale 1/sqrt(8) post-hoc ----
  {
    const float* efr = ef + (ebase + l16) * 8;
    v2f a0 = {efr[koff], efr[koff + 1]};
    v2f a1 = {efr[4 + koff], efr[4 + koff + 1]};
    v8f acc[4] = {vzero, vzero, vzero, vzero};
#pragma unroll
    for (int t = 0; t < 4; ++t) {
      const int n = t * 16 + l16;
      v2f b0 = {w0[koff * 64 + n], w0[(koff + 1) * 64 + n]};
      v2f b1 = {w0[(4 + koff) * 64 + n], w0[(5 + koff) * 64 + n]};
      acc[t] = wmma_f32(a0, b0, acc[t]);
      acc[t] = wmma_f32(a1, b1, acc[t]);
    }
#pragma unroll
    for (int t = 0; t < 4; ++t)
#pragma unroll
      for (int j = 0; j < 8; ++j)
        hA[(j + halfw * 8) * HS + t * 16 + l16] =
            act_silu(acc[t][j] * 0.35355339059327373f);
  }
  __syncthreads();

  // ---- MLP layers 1,2: (16x64) @ (64x64), scale 1/sqrt(64) ----
#pragma unroll
  for (int l = 0; l < 2; ++l) {
    const float* W   = (l == 0) ? w1 : w2;
    const float* src = (l == 0) ? hA : hB;
    float* dst       = (l == 0) ? hB : hA;
    v8f acc[4] = {vzero, vzero, vzero, vzero};
#pragma unroll 4
    for (int k = 0; k < 16; ++k) {
      const int kk = k * 4 + koff;
      v2f a = {src[l16 * HS + kk], src[l16 * HS + kk + 1]};
#pragma unroll
      for (int t = 0; t < 4; ++t) {
        const int n = t * 16 + l16;
        v2f b = {W[kk * 64 + n], W[(kk + 1) * 64 + n]};
        acc[t] = wmma_f32(a, b, acc[t]);
      }
    }
#pragma unroll
    for (int t = 0; t < 4; ++t)
#pragma unroll
      for (int j = 0; j < 8; ++j)
        dst[(j + halfw * 8) * HS + t * 16 + l16] = act_silu(acc[t][j] * 0.125f);
    __syncthreads();
  }
  // h now lives in hA

  // ---- layer 3: tpw = h @ (w3/sqrt(64)), fused with gather + tensor product.
  // tpw cols: [0,128)=w00 -> mid_s[u]; [128,256)=w01 -> mid_v[i][u];
  //           [256,384)=w10 -> mid_v[i][128+u]; [384,512)=w11 -> mid_s[128+u]
  for (int g = 0; g < 8; ++g) {
    v8f acc[4] = {vzero, vzero, vzero, vzero};
#pragma unroll 4
    for (int k = 0; k < 16; ++k) {
      const int kk = k * 4 + koff;
      v2f a = {hA[l16 * HS + kk], hA[l16 * HS + kk + 1]};
#pragma unroll
      for (int t = 0; t < 4; ++t) {
        const int n = (g * 4 + t) * 16 + l16;
        v2f b = {w3[kk * 512 + n], w3[(kk + 1) * 512 + n]};
        acc[t] = wmma_f32(a, b, acc[t]);
      }
    }
#pragma unroll
    for (int t = 0; t < 4; ++t) {
      const int ti  = g * 4 + t;
      const int n   = ti * 16 + l16;
      const int cls = n >> 7;  // uniform across the wave for a given tile
      const int u   = n & 127;
#pragma unroll
      for (int j = 0; j < 8; ++j) {
        const int m   = j + halfw * 8;
        const float v = acc[t][j] * 0.125f;  // / sqrt(64)
        const float* nr = wsn + ssd[m] * 512;  // gathered node row (L2-hot)
        const float sh0 = ssh[m * 4 + 0];
        float* mrow = mid + m * MS;
        if (cls == 0) {  // a0 = sqrt(.5) * w00 * ss * sh0
          mrow[u] = 0.7071067811865476f * v * nr[u] * sh0;
        } else if (cls == 1) {  // a1 = sqrt(1.5)/sqrt(3) * w01 * ss * sh1_i
          const float tv = 0.7071067811865476f * v * nr[u];
          mrow[256 + u] = tv * ssh[m * 4 + 1];
          mrow[512 + u] = tv * ssh[m * 4 + 2];
          mrow[768 + u] = tv * ssh[m * 4 + 3];
        } else if (cls == 2) {  // b1 = sqrt(1.5)/sqrt(3) * w10 * sh0 * vs_i
          const float tv = 0.7071067811865476f * v * sh0;
          mrow[384 + u] = tv * nr[128 + u];
          mrow[640 + u] = tv * nr[256 + u];
          mrow[896 + u] = tv * nr[384 + u];
        } else {  // b0 = sqrt(.5)/sqrt(3) * w11 * dot(vs, sh1)
          const float dot = nr[128 + u] * ssh[m * 4 + 1] +
                            nr[256 + u] * ssh[m * 4 + 2] +
                            nr[384 + u] * ssh[m * 4 + 3];
          mrow[128 + u] = 0.4082482904638631f * v * dot;
        }
      }
    }
  }
  __syncthreads();

  // ---- output GEMMs: out_s = mid_s @ Wos / 16; out_v_i = mid_v_i @ Wov / 16
  for (int ng = 0; ng < 8; ++ng) {
    v8f aS = vzero, aV0 = vzero, aV1 = vzero, aV2 = vzero;
    const int n = ng * 16 + l16;
    const float* mrow = mid + l16 * MS;
#pragma unroll 2
    for (int k = 0; k < 64; ++k) {
      const int kk = k * 4 + koff;
      v2f xs = {mrow[kk], mrow[kk + 1]};
      v2f x0 = {mrow[256 + kk], mrow[256 + kk + 1]};
      v2f x1 = {mrow[512 + kk], mrow[512 + kk + 1]};
      v2f x2 = {mrow[768 + kk], mrow[768 + kk + 1]};
      v2f bs = {Wos[kk * 128 + n], Wos[(kk + 1) * 128 + n]};
      v2f bv = {Wov[kk * 128 + n], Wov[(kk + 1) * 128 + n]};
      aS  = wmma_f32(xs, bs, aS);   // 4 independent chains hide WMMA latency
      aV0 = wmma_f32(x0, bv, aV0);
      aV1 = wmma_f32(x1, bv, aV1);
      aV2 = wmma_f32(x2, bv, aV2);
    }
#pragma unroll
    for (int j = 0; j < 8; ++j) {
      const int e = ebase + j + halfw * 8;
      float* orow = out + e * 512;
      orow[n] = aS[j] * 0.0625f;  // out layout: [out_s(128) | out_v (u,i) interleaved]
      orow[128 + 3 * n + 0] = aV0[j] * 0.0625f;
      orow[128 + 3 * n + 1] = aV1[j] * 0.0625f;
      orow[128 + 3 * n + 2] = aV2[j] * 0.0625f;
    }
  }
}

// ---------------------------------------------------------------------------
extern "C" void kernel_launch(void* const* d_in, const int* in_sizes, int n_in,
                              void* d_out, int out_size, void* d_ws,
                              size_t ws_size, hipStream_t stream) {
  (void)in_sizes; (void)n_in; (void)out_size; (void)ws_size;
  const float* node_feats = (const float*)d_in[0];
  const float* edge_attrs = (const float*)d_in[1];
  const float* edge_feats = (const float*)d_in[2];
  const int*   edge_index = (const int*)d_in[3];  // row 0 = sender
  const float* W_up_s = (const float*)d_in[4];
  const float* W_up_v = (const float*)d_in[5];
  const float* mlp_w0 = (const float*)d_in[6];
  const float* mlp_w1 = (const float*)d_in[7];
  const float* mlp_w2 = (const float*)d_in[8];
  const float* mlp_w3 = (const float*)d_in[9];
  const float* W_out_s = (const float*)d_in[10];
  const float* W_out_v = (const float*)d_in[11];
  float* out = (float*)d_out;
  float* wsn = (float*)d_ws;  // 20000 x 512 f32 = 41 MB (fits in 192 MB L2)

  // Kernel 1: 1250 node tiles, 8 waves per block
  mace_node_up<<<(1250 + 7) / 8, 256, 0, stream>>>(node_feats, W_up_s, W_up_v,
                                                   wsn);

  // Kernel 2: 6250 edge tiles, one wave per block, ~73 KB dynamic LDS
  const size_t shmem = (size_t)(32 * 68 + 16 * 1028 + 64 + 16) * sizeof(float);
  mace_edge<<<6250, 32, shmem, stream>>>(edge_attrs, edge_feats, edge_index,
                                         mlp_w0, mlp_w1, mlp_w2, mlp_w3,
                                         W_out_s, W_out_v, wsn, out);
}